// KMaxTensorPooling_87067577025516
// MI455X (gfx1250) — compile-verified
//
#include <hip/hip_runtime.h>
#include <hip/hip_bf16.h>

typedef __attribute__((ext_vector_type(2))) float v2f;
typedef __attribute__((ext_vector_type(8))) float v8f;

#define BATCH   64
#define NPTS    8192
#define DDIM    256
#define KTOP    128

// ---------------------------------------------------------------------------
// Phase 1: per-row sum-of-squares via V_WMMA_F32_16X16X4_F32.
// One wave handles 16 rows x 256 cols. A = x^2 tile (16x4), B = ones(4x16),
// accumulated over 64 K-steps. Accumulator row r = norm(row_base + r)
// (replicated across N), rows 8..15 live in lanes 16..31.
// ---------------------------------------------------------------------------
__global__ void __launch_bounds__(256)
kmax_norms_wmma_kernel(const float* __restrict__ x, float* __restrict__ norms,
                       long total_rows) {
    const int wave = (int)((blockIdx.x * blockDim.x + threadIdx.x) >> 5);
    const int lane = (int)(threadIdx.x & 31);
    const long row_base = (long)wave * 16;
    if (row_base >= total_rows) return;            // wave-uniform: EXEC stays all-ones

    const int half = lane >> 4;                    // 0: K={0,1}, 1: K={2,3}
    const int m    = lane & 15;                    // row within tile (A matrix M)
    const float* __restrict__ rowp = x + (row_base + m) * (long)DDIM + 2 * half;

    v8f acc = {};
    v2f ones; ones.x = 1.0f; ones.y = 1.0f;        // B = all-ones -> row sums

    #pragma unroll 4
    for (int k0 = 0; k0 < DDIM; k0 += 4) {
        v2f a = *(const v2f*)(rowp + k0);
        a.x *= a.x;
        a.y *= a.y;
        acc = __builtin_amdgcn_wmma_f32_16x16x4_f32(
            /*neg_a=*/false, a, /*neg_b=*/false, ones,
            /*c_mod=*/(short)0, acc, /*reuse_a=*/false, /*reuse_b=*/false);
    }

    // Lane 0 holds norms of rows 0..7 in acc[0..7]; lane 16 holds rows 8..15.
    if (m == 0) {
        float* np = norms + row_base + half * 8;
        #pragma unroll
        for (int r = 0; r < 8; ++r) np[r] = acc[r];
    }
}

// ---------------------------------------------------------------------------
// Phase 2+3: per-batch top-128 by norm (exact jax top_k order) + fused gather.
// Keys: (norm_bits << 13) | (NPTS-1-idx). Norms are >= 0 so f32 bit patterns
// compare monotonically as unsigned. Descending key sort == descending norm,
// ascending index on ties. Full bitonic sort of 8192 u64 keys in 64 KB LDS.
// ---------------------------------------------------------------------------
__global__ void __launch_bounds__(1024, 1)
kmax_topk_gather_kernel(const float* __restrict__ x,
                        const float* __restrict__ norms,
                        float* __restrict__ out) {
    __shared__ unsigned long long keys[NPTS];

    const int b   = blockIdx.x;
    const int tid = (int)threadIdx.x;
    const int nthreads = (int)blockDim.x;

    const float* __restrict__ nb = norms + (long)b * NPTS;
    for (int i = tid; i < NPTS; i += nthreads) {
        unsigned int bits = __float_as_uint(nb[i]);
        keys[i] = ((unsigned long long)bits << 13) |
                  (unsigned long long)(NPTS - 1 - i);
    }
    __syncthreads();

    // Bitonic sort, descending.
    for (int k = 2; k <= NPTS; k <<= 1) {
        for (int j = k >> 1; j > 0; j >>= 1) {
            for (int i = tid; i < NPTS; i += nthreads) {
                int l = i ^ j;
                if (l > i) {
                    unsigned long long a = keys[i];
                    unsigned long long c = keys[l];
                    bool swap = ((i & k) == 0) ? (a < c) : (a > c);
                    if (swap) { keys[i] = c; keys[l] = a; }
                }
            }
            __syncthreads();
        }
    }

    // Fused gather of the top-128 rows, float4 (16B) coalesced copies.
    const float4* __restrict__ xb =
        reinterpret_cast<const float4*>(x + (long)b * NPTS * DDIM);
    float4* __restrict__ ob =
        reinterpret_cast<float4*>(out + (long)b * KTOP * DDIM);

    const int vec_per_row = DDIM / 4;  // 64
    for (int e = tid; e < KTOP * vec_per_row; e += nthreads) {
        int j   = e >> 6;              // output rank 0..127
        int d4  = e & (vec_per_row - 1);
        int idx = (NPTS - 1) - (int)(keys[j] & 0x1FFFull);
        ob[(long)j * vec_per_row + d4] = xb[(long)idx * vec_per_row + d4];
    }
}

// ---------------------------------------------------------------------------
extern "C" void kernel_launch(void* const* d_in, const int* in_sizes, int n_in,
                              void* d_out, int out_size, void* d_ws, size_t ws_size,
                              hipStream_t stream) {
    const float* x   = (const float*)d_in[0];        // [64, 8192, 256] f32
    float*       out = (float*)d_out;                // [64, 128, 256] f32
    float*       norms = (float*)d_ws;               // [64, 8192] f32 (2 MB)

    const long total_rows = (long)BATCH * NPTS;      // 524288, 16 rows/wave
    const int  block1 = 256;                         // 8 waves -> 128 rows/block
    const int  grid1  = (int)(total_rows / 128);     // 4096 blocks, no ragged waves

    kmax_norms_wmma_kernel<<<grid1, block1, 0, stream>>>(x, norms, total_rows);
    kmax_topk_gather_kernel<<<BATCH, 1024, 0, stream>>>(x, norms, out);
}